// YOLOV7TorchObjectDetector_80599356277071
// MI455X (gfx1250) — compile-verified
//
#include <hip/hip_runtime.h>

// ---------------- problem constants (from reference) ----------------
constexpr int B_      = 32;
constexpr int N_      = 25200;
constexpr int NC_     = 80;
constexpr int TOPK_   = 1000;
constexpr int MAXDET_ = 300;
constexpr float CONF_THR = 0.45f;
constexpr float IOU_THR  = 0.45f;

// output layout: det (B,300,6) | logits (B,300,80) | valid (B,300) as float
constexpr size_t DET_OFF = 0;
constexpr size_t LOG_OFF = (size_t)B_ * MAXDET_ * 6;                 // 57600
constexpr size_t VAL_OFF = LOG_OFF + (size_t)B_ * MAXDET_ * NC_;     // 825600

// ---------------- vector typedefs ----------------
typedef float    v2f  __attribute__((ext_vector_type(2)));
typedef float    v8f  __attribute__((ext_vector_type(8)));
typedef unsigned uint32x4 __attribute__((ext_vector_type(4)));
typedef int      int32x4  __attribute__((ext_vector_type(4)));
typedef int      int32x8  __attribute__((ext_vector_type(8)));

#if __has_include(<hip/amd_detail/amd_gfx1250_TDM.h>)
#define TDM_6ARG 1
#endif

// =====================================================================
// Kernel A: stream pred tiles into LDS via the Tensor Data Mover, then
// compute conf = max_c(cls_c * obj), argmax, candidate key.
// grid = (ceil(N/256), B), block = 256, dynLDS = 256*85*4 bytes.
// =====================================================================
constexpr int ROWS_ = 256;

__global__ __launch_bounds__(256)
void score_kernel(const float* __restrict__ pred,
                  float* __restrict__ key, int* __restrict__ cls)
{
    extern __shared__ float smem[];
    const int b     = blockIdx.y;
    const int rows0 = blockIdx.x * ROWS_;
    const int t     = threadIdx.x;

    // One wave issues the TDM DMA: 2D tile (85 x up-to-256 rows of 4B data),
    // tensor_dim1 clips the tail block (OOB reads return zero).
    if (t < 32) {
        unsigned long long gaddr =
            (unsigned long long)(pred + ((size_t)b * N_ + rows0) * 85);
        unsigned td1 = (unsigned)(N_ - rows0);
        // D# group0: count=1, lds_addr=0, 57-bit global addr, type=2
        uint32x4 g0 = { 1u,
                        0u,
                        (unsigned)(gaddr & 0xFFFFFFFFu),
                        (unsigned)(((gaddr >> 32) & 0x01FFFFFFu) | 0x80000000u) };
        // D# group1: data_size=4B, tensor_dim0=85, tensor_dim1=td1,
        // tile_dim0=85, tile_dim1=256, tensor_dim0_stride=85
        int32x8 g1 = { (int)(2u << 16),                         // data_size=2 (4B)
                       (int)(85u << 16),                        // tensor_dim0 lo16
                       (int)((td1 & 0xFFFFu) << 16),            // dim0 hi16 | dim1 lo16
                       (int)((td1 >> 16) | (85u << 16)),        // dim1 hi16 | tile_dim0
                       (int)(unsigned)ROWS_,                    // tile_dim1 | tile_dim2=0
                       (int)85,                                 // dim0_stride lo32
                       0, 0 };
        int32x4 gz = { 0, 0, 0, 0 };
#ifdef TDM_6ARG
        int32x8 gz8 = { 0, 0, 0, 0, 0, 0, 0, 0 };
        __builtin_amdgcn_tensor_load_to_lds(g0, g1, gz, gz, gz8, 0);
#else
        __builtin_amdgcn_tensor_load_to_lds(g0, g1, gz, gz, 0);
#endif
        __builtin_amdgcn_s_wait_tensorcnt(0);
    }
    __syncthreads();

    const int n = rows0 + t;
    if (n < N_) {
        const float* row = smem + t * 85;
        float obj  = row[4];
        float best = -1e30f;
        int   bi   = 0;
#pragma unroll 4
        for (int c = 0; c < NC_; ++c) {
            float s = row[5 + c] * obj;
            if (s > best) { best = s; bi = c; }   // first-max (matches argmax)
        }
        bool cand = (obj > CONF_THR) && (best > CONF_THR);
        key[(size_t)b * N_ + n] = cand ? best : -1.0f;
        cls[(size_t)b * N_ + n] = bi;
    }
}

// monotonic float->uint mapping (total order, works for -1.0 fill values)
__device__ inline unsigned mapf(float f)
{
    unsigned u = __float_as_uint(f);
    return (u & 0x80000000u) ? ~u : (u | 0x80000000u);
}

// =====================================================================
// Kernel B: per-batch radix-select (4x8-bit passes) for the 1000th
// largest key, compact winners, bitonic-sort 1024 entries descending
// (conf desc, idx asc tie-break). grid = B, block = 256.
// =====================================================================
__global__ __launch_bounds__(256)
void topk_kernel(const float* __restrict__ key, const int* __restrict__ cls,
                 float* __restrict__ selConf, int* __restrict__ selIdx,
                 int* __restrict__ selCls)
{
    __shared__ int      hist[256];
    __shared__ unsigned sh_pref;
    __shared__ int      sh_k;
    __shared__ int      cntHi, cntTie;
    __shared__ float    sconf[1024];
    __shared__ int      sidx[1024];
    __shared__ int      scls[1024];

    const int b = blockIdx.x;
    const int t = threadIdx.x;
    const int bd = blockDim.x;
    const float* kb = key + (size_t)b * N_;
    const int*   cb = cls + (size_t)b * N_;

    unsigned pref = 0;
    int k = TOPK_;
    for (int shift = 24; shift >= 0; shift -= 8) {
        for (int i = t; i < 256; i += bd) hist[i] = 0;
        __syncthreads();
        unsigned prefmask = (shift == 24) ? 0u : (0xFFFFFFFFu << (shift + 8));
        for (int n = t; n < N_; n += bd) {
            __builtin_prefetch(kb + n + 4096, 0, 1);   // global_prefetch_b8
            unsigned u = mapf(kb[n]);
            if ((u & prefmask) == pref)
                atomicAdd(&hist[(u >> shift) & 255], 1);
        }
        __syncthreads();
        if (t == 0) {
            int cum = 0, bin = 255;
            for (; bin > 0; --bin) {
                if (cum + hist[bin] >= k) break;
                cum += hist[bin];
            }
            sh_pref = pref | (((unsigned)bin) << shift);
            sh_k    = k - cum;
        }
        __syncthreads();
        pref = sh_pref;
        k    = sh_k;
        __syncthreads();
    }
    const unsigned T = pref;   // value of the 1000th-largest mapped key

    // init sort arrays with sentinels
    for (int i = t; i < 1024; i += bd) {
        sconf[i] = -3.0e38f;
        sidx[i]  = 0x7FFFFFFF;
        scls[i]  = 0;
    }
    if (t == 0) { cntHi = 0; cntTie = 0; }
    __syncthreads();

    for (int n = t; n < N_; n += bd) {
        unsigned u = mapf(kb[n]);
        if (u > T) {
            int p = atomicAdd(&cntHi, 1);
            sconf[p] = kb[n]; sidx[p] = n; scls[p] = cb[n];
        }
    }
    __syncthreads();
    const int m = cntHi;           // strictly-greater count (< TOPK)
    for (int n = t; n < N_; n += bd) {
        unsigned u = mapf(kb[n]);
        if (u == T) {
            int p = m + atomicAdd(&cntTie, 1);
            if (p < TOPK_) { sconf[p] = kb[n]; sidx[p] = n; scls[p] = cb[n]; }
        }
    }
    __syncthreads();

    // bitonic sort, 1024 elems, descending by (conf, then smaller idx first)
    for (unsigned kk = 2; kk <= 1024; kk <<= 1) {
        for (unsigned j = kk >> 1; j > 0; j >>= 1) {
            __syncthreads();
            for (unsigned i = t; i < 1024; i += bd) {
                unsigned ixj = i ^ j;
                if (ixj > i) {
                    float a = sconf[i], c = sconf[ixj];
                    int ia = sidx[i], ic = sidx[ixj];
                    bool aFirst = (a > c) || (a == c && ia < ic);
                    bool desc = ((i & kk) == 0);
                    if (desc ? !aFirst : aFirst) {
                        sconf[i] = c; sconf[ixj] = a;
                        sidx[i] = ic; sidx[ixj] = ia;
                        int tc = scls[i]; scls[i] = scls[ixj]; scls[ixj] = tc;
                    }
                }
            }
        }
    }
    __syncthreads();

    for (int i = t; i < TOPK_; i += bd) {
        selConf[(size_t)b * TOPK_ + i] = sconf[i];
        selIdx [(size_t)b * TOPK_ + i] = sidx[i];
        selCls [(size_t)b * TOPK_ + i] = scls[i];
    }
}

// =====================================================================
// Kernel C: per-batch — gather xywh, WMMA f32 16x16x4 xywh->xyxy
// transform, 1000x1000 IoU bitmask, wave32 greedy NMS, rank/compact
// top-300, write det/logits/valid. grid = B, block = 1024.
// =====================================================================
__global__ __launch_bounds__(1024)
void nms_kernel(const float* __restrict__ pred, const float* __restrict__ logits,
                const float* __restrict__ selConf, const int* __restrict__ selIdx,
                const int* __restrict__ selCls, unsigned* __restrict__ supp,
                float* __restrict__ out)
{
    __shared__ float    sx [1024 * 4];   // xywh (padded rows zero)
    __shared__ float    sxy[1024 * 4];   // xyxy
    __shared__ float    sarea[1024];
    __shared__ float    sconf[1024];
    __shared__ int      sidx[1024];
    __shared__ unsigned skeep0[32], skeep[32];
    __shared__ int      swpre[33];
    __shared__ int      somap[MAXDET_];
    __shared__ int      sgidx[MAXDET_];

    const int b = blockIdx.x;
    const int t = threadIdx.x;
    unsigned* suppb = supp + (size_t)b * TOPK_ * 32;

    // gather selected boxes
    if (t < TOPK_) {
        int g = selIdx[(size_t)b * TOPK_ + t];
        const float* p = pred + ((size_t)b * N_ + g) * 85;
        sx[t * 4 + 0] = p[0]; sx[t * 4 + 1] = p[1];
        sx[t * 4 + 2] = p[2]; sx[t * 4 + 3] = p[3];
        sconf[t] = selConf[(size_t)b * TOPK_ + t];
        sidx[t]  = g;
    } else {
        sx[t * 4 + 0] = 0.f; sx[t * 4 + 1] = 0.f;
        sx[t * 4 + 2] = 0.f; sx[t * 4 + 3] = 0.f;
        sconf[t] = -1.f; sidx[t] = 0;
    }
    if (t < 32) skeep0[t] = 0u;
    __syncthreads();
    if (t < TOPK_ && sconf[t] > 0.f)
        atomicOr(&skeep0[t >> 5], 1u << (t & 31));
    __syncthreads();

    // ---- WMMA xywh -> xyxy:  D(16x16) = A(16x4 boxes) x B(4x16 transform) ----
    // B columns 0..3 encode [cx-w/2, cy-h/2, cx+w/2, cy+h/2]; cols 4..15 = 0.
    if (t < 32) {
        const int lane = t;
        // B layout guess mirrors C/D striping: VGPR0 = rows K0(l<16)/K2(l>=16),
        // VGPR1 = rows K1/K3, N = lane%16.
        float b0, b1;
        if (lane < 16) {
            b0 = (lane == 0 || lane == 2) ? 1.f : 0.f;          // K=0 row
            b1 = (lane == 1 || lane == 3) ? 1.f : 0.f;          // K=1 row
        } else {
            int n = lane - 16;
            b0 = (n == 0) ? -0.5f : (n == 2) ? 0.5f : 0.f;      // K=2 row
            b1 = (n == 1) ? -0.5f : (n == 3) ? 0.5f : 0.f;      // K=3 row
        }
        v2f bm; bm.x = b0; bm.y = b1;
        for (int tile = 0; tile < 64; ++tile) {
            int mrow = tile * 16 + (lane & 15);
            int kcol = (lane >> 4) * 2;
            v2f a; a.x = sx[mrow * 4 + kcol]; a.y = sx[mrow * 4 + kcol + 1];
            v8f c = {};
            c = __builtin_amdgcn_wmma_f32_16x16x4_f32(
                    false, a, false, bm, (short)0, c, false, false);
            int comp = lane & 15;
            if (comp < 4) {
                int rbase = tile * 16 + (lane >> 4) * 8;
#pragma unroll
                for (int v = 0; v < 8; ++v)
                    sxy[(rbase + v) * 4 + comp] = c[v];
            }
        }
    }
    __syncthreads();

    sarea[t] = (sxy[t * 4 + 2] - sxy[t * 4 + 0]) *
               (sxy[t * 4 + 3] - sxy[t * 4 + 1]);
    __syncthreads();

    // ---- suppression bitmask: row i, word w covers j = 32w..32w+31 (j>i only)
    for (int wi = t; wi < TOPK_ * 32; wi += 1024) {
        int i = wi >> 5, w = wi & 31;
        float xi1 = sxy[i * 4 + 0], yi1 = sxy[i * 4 + 1];
        float xi2 = sxy[i * 4 + 2], yi2 = sxy[i * 4 + 3];
        float ai  = sarea[i];
        unsigned bits = 0;
        int j0 = w * 32;
#pragma unroll 4
        for (int jj = 0; jj < 32; ++jj) {
            int j = j0 + jj;
            if (j <= i || j >= TOPK_) continue;
            float xl = fmaxf(xi1, sxy[j * 4 + 0]);
            float yt = fmaxf(yi1, sxy[j * 4 + 1]);
            float xr = fminf(xi2, sxy[j * 4 + 2]);
            float yb = fminf(yi2, sxy[j * 4 + 3]);
            float w_ = fmaxf(xr - xl, 0.f), h_ = fmaxf(yb - yt, 0.f);
            float inter = w_ * h_;
            float iou = inter / (ai + sarea[j] - inter + 1e-9f);
            if (iou > IOU_THR) bits |= (1u << jj);
        }
        suppb[i * 32 + w] = bits;
    }
    __threadfence_block();
    __syncthreads();

    // ---- greedy NMS: one wave, keep-word per lane, 1000 sequential steps
    if (t < 32) {
        unsigned kw = skeep0[t];
        for (int i = 0; i < TOPK_; ++i) {
            unsigned word = (unsigned)__shfl((int)kw, i >> 5, 32);
            if ((word >> (i & 31)) & 1u)
                kw &= ~suppb[i * 32 + t];
        }
        skeep[t] = kw;
    }
    __syncthreads();

    if (t == 0) {
        int c = 0;
        for (int w = 0; w < 32; ++w) { swpre[w] = c; c += __popc(skeep[w]); }
        swpre[32] = c;
    }
    if (t < MAXDET_) somap[t] = -1;
    __syncthreads();

    if (t < TOPK_) {
        unsigned word = skeep[t >> 5];
        if ((word >> (t & 31)) & 1u) {
            int rank = swpre[t >> 5] +
                       __popc(word & ((1u << (t & 31)) - 1u));
            if (rank < MAXDET_) somap[rank] = t;
        }
    }
    __syncthreads();

    // ---- det + valid
    if (t < MAXDET_) {
        int j = somap[t];
        sgidx[t] = (j >= 0) ? sidx[j] : 0;
        float* det = out + DET_OFF + ((size_t)b * MAXDET_ + t) * 6;
        float* vo  = out + VAL_OFF + (size_t)b * MAXDET_ + t;
        if (j >= 0) {
            det[0] = sxy[j * 4 + 0]; det[1] = sxy[j * 4 + 1];
            det[2] = sxy[j * 4 + 2]; det[3] = sxy[j * 4 + 3];
            det[4] = sconf[j];
            det[5] = (float)selCls[(size_t)b * TOPK_ + j];
            *vo = 1.f;
        } else {
            det[0] = det[1] = det[2] = det[3] = det[4] = det[5] = 0.f;
            *vo = 0.f;
        }
    }
    __syncthreads();

    // ---- logits gather (zeros for invalid rows)
    const float* lg = logits + (size_t)b * N_ * NC_;
    float* lo = out + LOG_OFF + (size_t)b * MAXDET_ * NC_;
    for (int e = t; e < MAXDET_ * NC_; e += 1024) {
        int r = e / NC_, c = e % NC_;
        int j = somap[r];
        lo[e] = (j >= 0) ? lg[(size_t)sgidx[r] * NC_ + c] : 0.f;
    }
}

// =====================================================================
extern "C" void kernel_launch(void* const* d_in, const int* in_sizes, int n_in,
                              void* d_out, int out_size, void* d_ws, size_t ws_size,
                              hipStream_t stream)
{
    const float* pred   = (const float*)d_in[0];   // (B,N,85) f32
    const float* logits = (const float*)d_in[1];   // (B,N,80) f32
    float* out = (float*)d_out;

    // workspace carve-up
    float*    key     = (float*)d_ws;                               // B*N
    int*      cls     = (int*)(key + (size_t)B_ * N_);              // B*N
    float*    selConf = (float*)(cls + (size_t)B_ * N_);            // B*TOPK
    int*      selIdx  = (int*)(selConf + (size_t)B_ * TOPK_);       // B*TOPK
    int*      selCls  = (int*)(selIdx + (size_t)B_ * TOPK_);        // B*TOPK
    unsigned* supp    = (unsigned*)(selCls + (size_t)B_ * TOPK_);   // B*TOPK*32

    dim3 gA((N_ + ROWS_ - 1) / ROWS_, B_);
    score_kernel<<<gA, ROWS_, ROWS_ * 85 * sizeof(float), stream>>>(pred, key, cls);
    topk_kernel<<<B_, 256, 0, stream>>>(key, cls, selConf, selIdx, selCls);
    nms_kernel<<<B_, 1024, 0, stream>>>(pred, logits, selConf, selIdx, selCls,
                                        supp, out);
}